// GCNNet_1056561954976
// MI455X (gfx1250) — compile-verified
//
#include <hip/hip_runtime.h>
#include <hip/hip_bf16.h>

typedef __attribute__((ext_vector_type(2))) float v2f;
typedef __attribute__((ext_vector_type(8))) float v8f;

namespace {
constexpr int kN0 = 200000, kN1 = 150000, kN2 = 120000, kN3 = 100000;
constexpr long kE0 = 2000000, kE1 = 1000000, kE2 = 500000, kESUB = 250000;
constexpr int kRelDim = 32;
}

// ---------------------------------------------------------------- utilities

__global__ void zero_f32(float* __restrict__ p, long n) {
  long i = (long)blockIdx.x * blockDim.x + threadIdx.x;
  long stride = (long)gridDim.x * blockDim.x;
  for (; i < n; i += stride) p[i] = 0.0f;
}

__global__ void count_deg(const int* __restrict__ src, const int* __restrict__ dst,
                          float* __restrict__ dego, float* __restrict__ degi,
                          long E, int slimit) {
  long e = (long)blockIdx.x * blockDim.x + threadIdx.x;
  if (e >= E) return;
  int s = src[e];
  if (s < slimit) atomicAdd(&dego[s], 1.0f);
  atomicAdd(&degi[dst[e]], 1.0f);
}

// mode 0: 1/max(deg,1)   mode 1: rsqrt(max(deg,1))
__global__ void make_scale(const float* __restrict__ deg, float* __restrict__ out,
                           int n, int mode) {
  int i = blockIdx.x * blockDim.x + threadIdx.x;
  if (i >= n) return;
  float d = fmaxf(deg[i], 1.0f);
  out[i] = mode ? rsqrtf(d) : (1.0f / d);
}

// agg[sidx[e]][f] += feat[gidx[e]][f] * (rs ? rs[gidx[e]] : 1)
__global__ void scatter_feat(const float* __restrict__ feat, int ldf,
                             const int* __restrict__ gidx, const int* __restrict__ sidx,
                             const float* __restrict__ rs,
                             float* __restrict__ agg, int ldagg,
                             long E, int fshift, int slimit) {
  long t = (long)blockIdx.x * blockDim.x + threadIdx.x;
  long total = E << fshift;
  if (t >= total) return;
  long e = t >> fshift;
  int f = (int)(t & ((1L << fshift) - 1));
  int s = sidx[e];
  if (s >= slimit) return;
  int g = gidx[e];
  float v = feat[(long)g * ldf + f];
  if (rs) v *= rs[g];
  atomicAdd(&agg[(long)s * ldagg + f], v);
}

// ------------------------------------------------------------- WMMA GEMM
// C[M x N] = post( A' @ W + bias ), W row-major K x N, fp32 WMMA 16x16x4.
// Each wave owns a 16(M) x 32(N) output slab: one A-frag feeds TWO independent
// WMMA accumulator chains (halves A traffic + loop overhead per matrix op).
// A' row r, col k:
//   gather mode (gidx0): k<KA ? A[gidx0[r]*lda+k] : A[gidx1[r]*lda+k-KA]
//   else               : A[r*lda+k] * (rowscale ? rowscale[r] : 1)
// post: +bias[col], optional relu, optional zero-row when rowmask[r]<0.5.
// M = gridDim.y*16 rows; block = wn waves covering 32*wn cols; K,N mult of 32/4.
// A rows must be 16B-aligned (all callers satisfy this).
__global__ __launch_bounds__(256) void wmma_gemm_f32(
    const float* __restrict__ A, int lda,
    const int* __restrict__ gidx0, const int* __restrict__ gidx1, int KA,
    const float* __restrict__ rowscale,
    const float* __restrict__ W,
    const float* __restrict__ bias,
    const float* __restrict__ rowmask,
    float* __restrict__ C, int ldc,
    int N, int K, int kshift, int relu) {
  extern __shared__ float As[];  // 16 * (K+1), odd stride => conflict-free
  const int tid = threadIdx.x;
  const int lane = tid & 31;
  const int wave = tid >> 5;
  const int nwaves = blockDim.x >> 5;
  const int row0 = blockIdx.y * 16;
  const int col0 = (blockIdx.x * nwaves + wave) * 32;
  const int lds = K + 1;

  // Stage 16 x K A-tile into LDS with float4 global loads
  // (gather + degree-scale fused here; scalar LDS stores tolerate padding).
  const int kv = K >> 2;  // float4 elements per row
  for (int idx = tid; idx < (kv << 4); idx += blockDim.x) {
    int r = idx >> (kshift - 2);
    int k4 = (idx & (kv - 1)) << 2;
    int grow = row0 + r;
    float4 v;
    if (gidx0) {
      int node = (k4 < KA) ? gidx0[grow] : gidx1[grow];
      int kk = (k4 < KA) ? k4 : (k4 - KA);
      v = *(const float4*)&A[(long)node * lda + kk];
    } else {
      v = *(const float4*)&A[(long)grow * lda + k4];
      if (rowscale) {
        float s = rowscale[grow];
        v.x *= s; v.y *= s; v.z *= s; v.w *= s;
      }
    }
    float* dst = &As[r * lds + k4];
    dst[0] = v.x; dst[1] = v.y; dst[2] = v.z; dst[3] = v.w;
  }
  __syncthreads();

  // ISA A-frag layout (32-bit A 16x4): lanes 0-15 hold K=k0,k0+1; lanes 16-31 K=k0+2,k0+3.
  const int lr = lane & 15;
  const bool hi = lane >= 16;
  const int kh = hi ? 2 : 0;
  const int colA = col0 + lr;       // tile 0 column
  const float* Arow = As + lr * lds;

  v8f acc0 = {};
  v8f acc1 = {};
  for (int k0 = 0; k0 < K; k0 += 4) {
    v2f a, b0, b1;
    a.x = Arow[k0 + kh];
    a.y = Arow[k0 + kh + 1];
    const float* Wr0 = W + (long)(k0 + kh) * N;
    const float* Wr1 = Wr0 + N;
    b0.x = Wr0[colA];       b0.y = Wr1[colA];        // 16 consecutive floats / half-wave
    b1.x = Wr0[colA + 16];  b1.y = Wr1[colA + 16];
    acc0 = __builtin_amdgcn_wmma_f32_16x16x4_f32(false, a, false, b0, (short)0, acc0, false, false);
    acc1 = __builtin_amdgcn_wmma_f32_16x16x4_f32(false, a, false, b1, (short)0, acc1, false, false);
  }

  // C layout: lanes 0-15 rows 0..7, lanes 16-31 rows 8..15; col = col0+lr (+16).
  const int rbase = hi ? 8 : 0;
  float bv0 = bias ? bias[colA] : 0.0f;
  float bv1 = bias ? bias[colA + 16] : 0.0f;
#pragma unroll
  for (int i = 0; i < 8; ++i) {
    int r = row0 + rbase + i;
    float v0 = acc0[i] + bv0;
    float v1 = acc1[i] + bv1;
    if (relu) { v0 = fmaxf(v0, 0.0f); v1 = fmaxf(v1, 0.0f); }
    if (rowmask && rowmask[r] < 0.5f) { v0 = 0.0f; v1 = 0.0f; }  // empty segment => 0
    C[(long)r * ldc + colA] = v0;
    C[(long)r * ldc + colA + 16] = v1;
  }
}

// ---------------------------------------------------------------- host

extern "C" void kernel_launch(void* const* d_in, const int* in_sizes, int n_in,
                              void* d_out, int out_size, void* d_ws, size_t ws_size,
                              hipStream_t stream) {
  (void)in_sizes; (void)n_in; (void)out_size; (void)ws_size;
  const int* e0_type = (const int*)d_in[0];
  const int* e0_src  = (const int*)d_in[1];
  const int* e0_dst  = (const int*)d_in[2];
  const int* e1_src  = (const int*)d_in[3];
  const int* e1_dst  = (const int*)d_in[4];
  const int* e2_src  = (const int*)d_in[5];
  const int* e2_dst  = (const int*)d_in[6];
  const int* sub_src = (const int*)d_in[7];
  const int* sub_dst = (const int*)d_in[8];
  const float* rel_emb = (const float*)d_in[9];
  const float* W_out = (const float*)d_in[10];
  const float* b_out = (const float*)d_in[11];
  const float* W_in  = (const float*)d_in[12];
  const float* b_in  = (const float*)d_in[13];
  const float* W1    = (const float*)d_in[14];
  const float* b1    = (const float*)d_in[15];
  const float* W2    = (const float*)d_in[16];
  const float* b2    = (const float*)d_in[17];
  const float* Wfc   = (const float*)d_in[18];
  const float* bfc   = (const float*)d_in[19];
  float* out = (float*)d_out;

  // workspace layout (fp32); every slab size is a multiple of 4 floats => 16B aligned
  float* ws = (float*)d_ws;
  size_t o = 0;
  auto take = [&](size_t n) { float* p = ws + o; o += n; return p; };
  float* s_out = take((size_t)kN1 * 32);
  float* s_in  = take((size_t)kN1 * 32);
  float* deg0o = take(kN1);
  float* deg0i = take(kN1);
  float* rs0o  = take(kN1);
  float* rs0i  = take(kN1);
  float* x     = take((size_t)kN1 * 64);
  float* deg1o = take(kN1);
  float* deg1i = take(kN2);
  float* rs1o  = take(kN1);
  float* rs1i  = take(kN2);
  float* agg1  = take((size_t)kN2 * 64);
  float* x2    = take((size_t)kN2 * 256);
  float* deg2o = take(kN2);
  float* deg2i = take(kN3);
  float* rs2o  = take(kN2);
  float* rs2i  = take(kN3);
  float* agg2  = take((size_t)kN3 * 256);
  float* x3    = take((size_t)kN3 * 256);

  auto zero = [&](float* p, long n) {
    zero_f32<<<2048, 256, 0, stream>>>(p, n);
  };
  auto gemm = [&](const float* A, int lda, const int* g0, const int* g1, int KA,
                  const float* rowscale, const float* W, const float* bias,
                  const float* rowmask, float* C, int ldc,
                  int M, int N, int K, int relu) {
    int kshift = 31 - __builtin_clz((unsigned)K);
    int wn = N / 32; if (wn > 8) wn = 8;      // waves per block, 32 cols each
    dim3 block(32 * wn);
    dim3 grid(N / (32 * wn), M / 16);
    size_t shmem = (size_t)16 * (K + 1) * sizeof(float);
    wmma_gemm_f32<<<grid, block, shmem, stream>>>(
        A, lda, g0, g1, KA, rowscale, W, bias, rowmask, C, ldc, N, K, kshift, relu);
  };

  // --- zero accumulators ---
  zero(s_out, (long)kN1 * 32); zero(s_in, (long)kN1 * 32);
  zero(deg0o, kN1); zero(deg0i, kN1);
  zero(deg1o, kN1); zero(deg1i, kN2); zero(agg1, (long)kN2 * 64);
  zero(deg2o, kN2); zero(deg2i, kN3); zero(agg2, (long)kN3 * 256);

  // --- Phase A: relation embedding mean-agg (linearity-folded), build x (N1 x 64) ---
  count_deg<<<(unsigned)((kE0 + 255) / 256), 256, 0, stream>>>(e0_src, e0_dst, deg0o, deg0i, kE0, kN1);
  {
    long tot = kE0 << 5;  // F = 32
    unsigned g = (unsigned)((tot + 255) / 256);
    scatter_feat<<<g, 256, 0, stream>>>(rel_emb, kRelDim, e0_type, e0_src, nullptr, s_out, 32, kE0, 5, kN1);
    scatter_feat<<<g, 256, 0, stream>>>(rel_emb, kRelDim, e0_type, e0_dst, nullptr, s_in, 32, kE0, 5, kN1);
  }
  make_scale<<<(kN1 + 255) / 256, 256, 0, stream>>>(deg0o, rs0o, kN1, 0);
  make_scale<<<(kN1 + 255) / 256, 256, 0, stream>>>(deg0i, rs0i, kN1, 0);
  // x[:, 0:32] = mean_out @ W_out (+ b_out where deg>0); x[:, 32:64] likewise
  gemm(s_out, 32, nullptr, nullptr, 0, rs0o, W_out, b_out, deg0o, x,      64, kN1, 32, 32, 0);
  gemm(s_in,  32, nullptr, nullptr, 0, rs0i, W_in,  b_in,  deg0i, x + 32, 64, kN1, 32, 32, 0);

  // --- Phase B: GCN layer 1 (N1 -> N2, F=64, H=256) ---
  count_deg<<<(unsigned)((kE1 + 255) / 256), 256, 0, stream>>>(e1_src, e1_dst, deg1o, deg1i, kE1, kN1);
  make_scale<<<(kN1 + 255) / 256, 256, 0, stream>>>(deg1o, rs1o, kN1, 1);
  make_scale<<<(kN2 + 255) / 256, 256, 0, stream>>>(deg1i, rs1i, kN2, 1);
  {
    long tot = kE1 << 6;  // F = 64
    scatter_feat<<<(unsigned)((tot + 255) / 256), 256, 0, stream>>>(
        x, 64, e1_src, e1_dst, rs1o, agg1, 64, kE1, 6, kN2);
  }
  gemm(agg1, 64, nullptr, nullptr, 0, rs1i, W1, b1, nullptr, x2, 256, kN2, 256, 64, 1);

  // --- Phase C: GCN layer 2 (N2 -> N3, F=256, H=256) ---
  count_deg<<<(unsigned)((kE2 + 255) / 256), 256, 0, stream>>>(e2_src, e2_dst, deg2o, deg2i, kE2, kN2);
  make_scale<<<(kN2 + 255) / 256, 256, 0, stream>>>(deg2o, rs2o, kN2, 1);
  make_scale<<<(kN3 + 255) / 256, 256, 0, stream>>>(deg2i, rs2i, kN3, 1);
  {
    long tot = kE2 << 8;  // F = 256
    scatter_feat<<<(unsigned)((tot + 255) / 256), 256, 0, stream>>>(
        x2, 256, e2_src, e2_dst, rs2o, agg2, 256, kE2, 8, kN3);
  }
  gemm(agg2, 256, nullptr, nullptr, 0, rs2i, W2, b2, nullptr, x3, 256, kN3, 256, 256, 1);

  // --- Phase D: edge head, concat-gather fused into the GEMM A-stage ---
  gemm(x3, 256, sub_src, sub_dst, 256, nullptr, Wfc, bfc, nullptr, out, 256,
       (int)kESUB, 256, 512, 0);
}